// InnerConvexViolationProjection_84000970375195
// MI455X (gfx1250) — compile-verified
//
#include <hip/hip_runtime.h>
#include <math.h>

// Problem constants (fixed by setup_inputs): B=16, S=32, m=n=256.
#define NPROB 512
#define N     256
#define K_IT  30
#define PWR_IT 5
#define LDA   260          // padded row stride in floats: 16B-aligned rows, conflict-free columns

typedef __attribute__((ext_vector_type(2))) float v2f;
typedef __attribute__((ext_vector_type(8))) float v8f;

// -------- block-wide reductions (wave32) --------
#define OP_SUM 0
#define OP_MAX 1
#define OP_MIN 2

__device__ __forceinline__ float red_op(float a, float b, int op) {
  return op == OP_SUM ? (a + b) : (op == OP_MAX ? fmaxf(a, b) : fminf(a, b));
}

__device__ __forceinline__ float blk_red(float v, float* sred, int op) {
  #pragma unroll
  for (int off = 16; off; off >>= 1)
    v = red_op(v, __shfl_xor(v, off, 32), op);
  const int w = threadIdx.x >> 5;
  const int ln = threadIdx.x & 31;
  if (ln == 0) sred[w] = v;
  __syncthreads();
  if (threadIdx.x < 8) {
    v = sred[threadIdx.x];
    #pragma unroll
    for (int off = 4; off; off >>= 1)
      v = red_op(v, __shfl_xor(v, off, 32), op);
    if (threadIdx.x == 0) sred[0] = v;
  }
  __syncthreads();
  v = sred[0];
  __syncthreads();
  return v;
}

// -------- WMMA matvec: (TRANS ? A^T : A) * f(vin), f = relu if RELU --------
// 8 waves; each wave owns 32 output rows as TWO independent 16-row accumulator
// chains in one K loop (2x WMMA ILP, B-frag shared by both chains).
// Epilogue (lanes 0/16 hold the results, D layout: VGPR j -> M=j / M=j+8):
//   EPI==0 : vout[i]  = acc
//   EPI==1 : vout[i]  = acc - bsub[i]          (fused  r = A x - b_tight)
//   EPI==2 : vout[i] -= eta * acc              (fused  x = x - eta * A^T relu(r))
template<bool TRANS, bool RELU, int EPI>
__device__ __forceinline__ void matvec256(const float* __restrict__ As,
                                          const float* __restrict__ vin,
                                          float* __restrict__ vout,
                                          const float* __restrict__ bsub,
                                          float eta) {
  const int tid = threadIdx.x;
  const int wid = tid >> 5;
  const int l   = tid & 31;
  const int mr  = l & 15;            // M index within 16-row block
  const int kh  = (l >> 4) << 1;     // K sub-offset: lanes 0-15 -> {0,1}, lanes 16-31 -> {2,3}
  const int rb0 = wid * 32;
  const int rb1 = rb0 + 16;

  v8f acc0 = {};
  v8f acc1 = {};
  #pragma unroll 4
  for (int kb = 0; kb < N; kb += 4) {
    const int kk = kb + kh;
    v2f a0, a1;
    if (!TRANS) {
      a0 = *(const v2f*)(As + (rb0 + mr) * LDA + kk);          // ds_load_b64, row-major
      a1 = *(const v2f*)(As + (rb1 + mr) * LDA + kk);
    } else {
      a0.x = As[kk * LDA + rb0 + mr];                          // column reads, lane-contiguous
      a0.y = As[(kk + 1) * LDA + rb0 + mr];
      a1.x = As[kk * LDA + rb1 + mr];
      a1.y = As[(kk + 1) * LDA + rb1 + mr];
    }
    v2f bv = *(const v2f*)(vin + kk);                          // vin[k] replicated over N cols
    if (RELU) { bv.x = fmaxf(bv.x, 0.f); bv.y = fmaxf(bv.y, 0.f); }
    acc0 = __builtin_amdgcn_wmma_f32_16x16x4_f32(
        false, a0, false, bv, (short)0, acc0, false, false);
    acc1 = __builtin_amdgcn_wmma_f32_16x16x4_f32(
        false, a1, false, bv, (short)0, acc1, false, false);
  }
  if (mr == 0) {
    const int half = (l >> 4) << 3;
    #pragma unroll
    for (int j = 0; j < 8; ++j) {
      const int i0 = rb0 + half + j;
      const int i1 = rb1 + half + j;
      if (EPI == 0) {
        vout[i0] = acc0[j];
        vout[i1] = acc1[j];
      } else if (EPI == 1) {
        vout[i0] = acc0[j] - bsub[i0];
        vout[i1] = acc1[j] - bsub[i1];
      } else {
        vout[i0] -= eta * acc0[j];
        vout[i1] -= eta * acc1[j];
      }
    }
  }
}

__global__ __launch_bounds__(256, 1)
void InnerConvexViolationProjection_84000970375195_kernel(
    const float* __restrict__ xg_all, const float* __restrict__ Ag_all,
    const float* __restrict__ bg_all, float* __restrict__ out_all) {
  extern __shared__ float smem[];
  float* As     = smem;                  // 256 * LDA
  float* bw     = As     + N * LDA;      // 256
  float* btight = bw     + N;            // 256
  float* vecX   = btight + N;            // 256 : current x
  float* vecX0  = vecX   + N;            // 256 : original x
  float* vecR   = vecX0  + N;            // 256 : r / Av / Ax
  float* vecG   = vecR   + N;            // 256 : v / d
  float* vecT   = vecG   + N;            // 256 : Ad
  float* sred   = vecT   + N;            // 16  : reduction scratch

  const int prob = blockIdx.x;
  const float*  Ag  = Ag_all + (size_t)prob * N * N;
  const float4* Ag4 = (const float4*)Ag;
  const float*  bg  = bg_all + (size_t)prob * N;
  const float*  xg  = xg_all + (size_t)prob * N;

  const int tid = threadIdx.x;
  const int wid = tid >> 5;
  const int l   = tid & 31;

  // ---- Load A into LDS (b128 global loads), row-normalize in-register, single LDS write ----
  for (int i = wid; i < N; i += 8) {
    float4 u0 = Ag4[i * 64 + l];         // 32 lanes * float4 = first 128 floats of row
    float4 u1 = Ag4[i * 64 + 32 + l];    // second 128 floats
    float s = u0.x * u0.x + u0.y * u0.y + u0.z * u0.z + u0.w * u0.w +
              u1.x * u1.x + u1.y * u1.y + u1.z * u1.z + u1.w * u1.w;
    #pragma unroll
    for (int off = 16; off; off >>= 1) s += __shfl_xor(s, off, 32);
    const float inv = 1.0f / fmaxf(sqrtf(s), 1e-12f);
    u0.x *= inv; u0.y *= inv; u0.z *= inv; u0.w *= inv;
    u1.x *= inv; u1.y *= inv; u1.z *= inv; u1.w *= inv;
    *(float4*)(As + i * LDA + 4 * l)       = u0;   // 16B-aligned LDS b128 stores
    *(float4*)(As + i * LDA + 128 + 4 * l) = u1;
    if (l == 0) {
      const float bwv = bg[i] * inv;
      bw[i] = bwv;
      btight[i] = bwv - 1e-3f;           // MU_INSIDE
    }
  }
  {
    const float xv = xg[tid];
    vecX[tid]  = xv;
    vecX0[tid] = xv;
    vecG[tid]  = 1.0f / 16.0f;           // v = ones / ||ones||, n = 256
  }
  __syncthreads();

  // ---- Power iteration: v <- normalize(A^T (A v)) ----
  for (int it = 0; it < PWR_IT; ++it) {
    matvec256<false, false, 0>(As, vecG, vecR, nullptr, 0.f); __syncthreads();
    matvec256<true,  false, 0>(As, vecR, vecG, nullptr, 0.f); __syncthreads();
    float g = vecG[tid];
    const float nrm = blk_red(g * g, sred, OP_SUM);
    vecG[tid] = g / (sqrtf(nrm) + 1e-12f);
    __syncthreads();
  }
  matvec256<false, false, 0>(As, vecG, vecR, nullptr, 0.f); __syncthreads();
  float av = vecR[tid];
  const float eta = 1.0f / (blk_red(av * av, sred, OP_SUM) + 1e-12f);  // RHO

  // ---- Stage 1: K-step UVP, fully fused: 2 barriers / iteration ----
  for (int it = 0; it < K_IT; ++it) {
    // vecR = A x - b_tight   (subtraction fused into epilogue)
    matvec256<false, false, 1>(As, vecX, vecR, btight, 0.f);
    __syncthreads();
    // vecX -= eta * A^T relu(vecR)   (update fused into epilogue; g never stored)
    matvec256<true, true, 2>(As, vecR, vecX, nullptr, eta);
    __syncthreads();
  }

  // ---- Stage 2: alpha-map toward original x, gated by feasibility ----
  matvec256<false, false, 0>(As, vecX, vecR, nullptr, 0.f); __syncthreads();   // Ax
  const float Axi  = vecR[tid];
  const float maxv = blk_red(fmaxf(Axi - bw[tid], 0.f), sred, OP_MAX);
  const bool  do_alpha = (maxv <= 1e-7f);                     // ALPHA_FEAS_TOL

  vecG[tid] = vecX0[tid] - vecX[tid];                         // d
  __syncthreads();
  matvec256<false, false, 0>(As, vecG, vecT, nullptr, 0.f); __syncthreads();   // Ad
  const float Adi   = vecT[tid];
  const float slack = bw[tid] - Axi;
  const float ai    = (Adi > 0.f) ? slack / (Adi + 1e-12f) : __builtin_inff();
  float alpha = blk_red(ai, sred, OP_MIN);
  if (!isfinite(alpha)) alpha = 1.0f;
  alpha = fminf(fmaxf(alpha - 1e-6f, 0.f), 1.0f);             // EPS_INSIDE, clip

  const float xu = vecX[tid];
  out_all[(size_t)prob * N + tid] = do_alpha ? (xu + alpha * vecG[tid]) : xu;
}

extern "C" void kernel_launch(void* const* d_in, const int* in_sizes, int n_in,
                              void* d_out, int out_size, void* d_ws, size_t ws_size,
                              hipStream_t stream) {
  (void)in_sizes; (void)n_in; (void)d_ws; (void)ws_size; (void)out_size;
  const float* xg = (const float*)d_in[0];
  const float* Ag = (const float*)d_in[1];
  const float* bg = (const float*)d_in[2];
  float* out = (float*)d_out;

  const size_t shmem = (size_t)(N * LDA + 7 * N + 16) * sizeof(float);  // ~267 KB < 320 KB/WGP
  hipFuncSetAttribute(
      (const void*)InnerConvexViolationProjection_84000970375195_kernel,
      hipFuncAttributeMaxDynamicSharedMemorySize, (int)shmem);

  InnerConvexViolationProjection_84000970375195_kernel
      <<<NPROB, 256, shmem, stream>>>(xg, Ag, bg, out);
}